// CNN_88098369175825
// MI455X (gfx1250) — compile-verified
//
#include <hip/hip_runtime.h>

typedef float v2f __attribute__((ext_vector_type(2)));
typedef float v8f __attribute__((ext_vector_type(8)));

#define VTH    0.2f
#define DECAY  0.2f
#define T_     2048
#define S_     256
#define NT     8
#define CH     64
#define CTOT   66
#define HID    10
#define EMB_STRIDE 2052   // 2048+4: halves of a wave hit disjoint bank sets
#define SPK_STRIDE 272    // 256+16: B-operand row pairs (k,k+2) 32 banks apart
#define G_STRIDE   260    // 256+4 : D rows (i, i+8) 32 banks apart

__global__ __launch_bounds__(256) void snn_fused_kernel(
    const float* __restrict__ x,      // (256,1,66,2048)
    const float* __restrict__ w_emb,  // (10,64)
    const float* __restrict__ b_emb,  // (10,)
    const float* __restrict__ w_ih,   // (40,10)
    const float* __restrict__ w_hh,   // (40,10)
    const float* __restrict__ b_ih,   // (40,)
    const float* __restrict__ b_hh,   // (40,)
    const float* __restrict__ w_cls,  // (2,10)
    const float* __restrict__ b_cls,  // (2,)
    float* __restrict__ out)          // (256,2)
{
    __shared__ float    embLDS[HID * EMB_STRIDE]; // 82,080 B
    __shared__ float    spkLDS[20 * SPK_STRIDE];  // 21,760 B  [espk(10); h(10)] x s
    __shared__ float    gLDS[48 * G_STRIDE];      // 49,920 B  gates x s (rows 40..47 pad)
    __shared__ float    biasG[40];
    __shared__ unsigned pooledCnt[HID * NT];

    const int tid  = threadIdx.x;
    const int b    = blockIdx.x;
    const int lane = tid & 31;
    const int wave = tid >> 5;
    const int half = lane >> 4;   // which 16-lane half of the wave
    const int lm   = lane & 15;

    // ---------------- setup ----------------
    if (tid < 40)        biasG[tid]     = b_ih[tid] + b_hh[tid];
    if (tid < HID * NT)  pooledCnt[tid] = 0u;

    // ---------------- Stage 1: emb GEMM via V_WMMA_F32_16X16X4_F32 -------
    // A = W_emb padded to 16x64 (f32).  16x4 f32 A layout:
    //   VGPR0: lanes0-15 -> K=k0,   lanes16-31 -> K=k0+2  (M = lane%16)
    //   VGPR1: lanes0-15 -> K=k0+1, lanes16-31 -> K=k0+3
    float a0[16], a1[16];
#pragma unroll
    for (int ks = 0; ks < 16; ++ks) {
        int c0 = ks * 4 + 2 * half;
        a0[ks] = (lm < HID) ? w_emb[lm * CH + c0]     : 0.0f;
        a1[ks] = (lm < HID) ? w_emb[lm * CH + c0 + 1] : 0.0f;
    }
    float biases[8];
#pragma unroll
    for (int i = 0; i < 8; ++i) {
        int hh = i + 8 * half;
        biases[i] = (hh < HID) ? b_emb[hh] : 0.0f;
    }

    // eeg base for this batch: channel 1 (skip audio ch 0)
    const float* xb = x + (size_t)b * CTOT * T_ + (size_t)1 * T_;

    // 8 waves x 16 tiles x 16 columns = 2048 columns
#pragma unroll 1
    for (int tile = 0; tile < 16; ++tile) {
        const int tau0 = wave * 256 + tile * 16;
        v8f acc = {0.f, 0.f, 0.f, 0.f, 0.f, 0.f, 0.f, 0.f};
        const float* p = xb + (size_t)(2 * half) * T_ + tau0 + lm;
#pragma unroll
        for (int ks = 0; ks < 16; ++ks) {
            float bb0 = p[(size_t)(4 * ks) * T_];
            float bb1 = p[(size_t)(4 * ks + 1) * T_];
            v2f av = {a0[ks], a1[ks]};
            v2f bv = {bb0, bb1};
            acc = __builtin_amdgcn_wmma_f32_16x16x4_f32(
                false, av, false, bv, (short)0, acc, false, false);
        }
        // D layout: acc[i] -> M = i + 8*half, N = tau0 + lm
        // rows {0,1,8,9} always valid; rows 2..7 only for half==0
#pragma unroll
        for (int i = 0; i < 2; ++i)
            embLDS[(i + 8 * half) * EMB_STRIDE + tau0 + lm] = acc[i] + biases[i];
        if (!half) {
#pragma unroll
            for (int i = 2; i < 8; ++i)
                embLDS[i * EMB_STRIDE + tau0 + lm] = acc[i] + biases[i];
        }
    }

    __syncthreads();

    // ---------------- Stage 2: spiking LSTM with WMMA gate GEMM ----------
    // Wcat[40][20] = [w_ih | w_hh]; G[40][256] = Wcat @ Spk per timestep.
    // Preload A operand for 3 row-tiles x 5 k-steps (K=20 exact).
    float wA0[3][5], wA1[3][5];
#pragma unroll
    for (int r = 0; r < 3; ++r) {
        int row = r * 16 + lm;
        bool vr = row < 40;
#pragma unroll
        for (int ks = 0; ks < 5; ++ks) {
            int col = ks * 4 + 2 * half;   // even, 0..18
            float w0 = 0.f, w1 = 0.f;
            if (vr) {
                w0 = (col < 10)     ? w_ih[row * 10 + col]      : w_hh[row * 10 + col - 10];
                w1 = (col + 1 < 10) ? w_ih[row * 10 + col + 1]  : w_hh[row * 10 + col - 9];
            }
            wA0[r][ks] = w0;
            wA1[r][ks] = w1;
        }
    }

    // t=0 LIF on emb: u0 = e, spike = e >= VTH; h_{-1} = 0
    float u[HID], cst[HID];
    unsigned espk = 0, hbits = 0;
#pragma unroll
    for (int h = 0; h < HID; ++h) {
        float e = embLDS[h * EMB_STRIDE + tid];
        u[h] = e; cst[h] = 0.f;
        if (e >= VTH) espk |= (1u << h);
    }
#pragma unroll
    for (int j = 0; j < HID; ++j) {
        spkLDS[j * SPK_STRIDE + tid]        = ((espk >> j) & 1u) ? 1.0f : 0.0f;
        spkLDS[(10 + j) * SPK_STRIDE + tid] = 0.0f;
    }

#pragma unroll 1
    for (int t = 0; t < NT; ++t) {
        __syncthreads();   // spikes visible -> gate GEMM

        // --- gate GEMM: each wave owns 2 column strips x 3 row tiles -----
#pragma unroll
        for (int sidx = 0; sidx < 2; ++sidx) {
            const int n0 = (2 * wave + sidx) * 16;
            float sb0[5], sb1[5];
#pragma unroll
            for (int ks = 0; ks < 5; ++ks) {
                int k0 = ks * 4 + 2 * half;
                sb0[ks] = spkLDS[k0 * SPK_STRIDE + n0 + lm];
                sb1[ks] = spkLDS[(k0 + 1) * SPK_STRIDE + n0 + lm];
            }
#pragma unroll
            for (int r = 0; r < 3; ++r) {
                v8f acc = {0.f, 0.f, 0.f, 0.f, 0.f, 0.f, 0.f, 0.f};
#pragma unroll
                for (int ks = 0; ks < 5; ++ks) {
                    v2f av = {wA0[r][ks], wA1[r][ks]};
                    v2f bv = {sb0[ks], sb1[ks]};
                    acc = __builtin_amdgcn_wmma_f32_16x16x4_f32(
                        false, av, false, bv, (short)0, acc, false, false);
                }
#pragma unroll
                for (int i = 0; i < 8; ++i)
                    gLDS[(r * 16 + i + 8 * half) * G_STRIDE + n0 + lm] = acc[i];
            }
        }

        __syncthreads();   // gates visible -> elementwise update

        // --- per-thread (s = tid) cell/hidden update ---------------------
        float gg[40];
#pragma unroll
        for (int k = 0; k < 40; ++k)
            gg[k] = gLDS[k * G_STRIDE + tid] + biasG[k];

        unsigned hnew = 0;
#pragma unroll
        for (int h = 0; h < HID; ++h) {
            float iv = gg[h], fv = gg[10 + h], gv = gg[20 + h], ov = gg[30 + h];
            float cn = (fv >= VTH ? cst[h] : 0.0f) +
                       ((iv >= VTH && gv >= VTH) ? 1.0f : 0.0f);
            cst[h] = cn;
            if (ov >= VTH && cn >= VTH) hnew |= (1u << h);
        }
        hbits = hnew;

        // pooled count: ballot + one LDS atomic per wave per (h,t)
#pragma unroll
        for (int h = 0; h < HID; ++h) {
            unsigned m = __builtin_amdgcn_ballot_w32((hbits >> h) & 1u);
            if (lane == 0 && m)
                atomicAdd(&pooledCnt[h * NT + t], (unsigned)__popc((int)m));
        }

        // LIF for next step + publish spikes for t+1
        if (t < NT - 1) {
            unsigned enew = 0;
#pragma unroll
            for (int h = 0; h < HID; ++h) {
                float e  = embLDS[h * EMB_STRIDE + (t + 1) * S_ + tid];
                float uu = ((espk >> h) & 1u) ? e : (u[h] * DECAY + e);
                u[h] = uu;
                if (uu >= VTH) enew |= (1u << h);
            }
            espk = enew;
#pragma unroll
            for (int j = 0; j < HID; ++j) {
                spkLDS[j * SPK_STRIDE + tid]        = ((espk  >> j) & 1u) ? 1.0f : 0.0f;
                spkLDS[(10 + j) * SPK_STRIDE + tid] = ((hbits >> j) & 1u) ? 1.0f : 0.0f;
            }
        }
    }

    __syncthreads();

    // ---------------- Stage 3: classifier + final LIF (2 threads) --------
    if (tid < 2) {
        const int o = tid;
        float wc[HID];
#pragma unroll
        for (int h = 0; h < HID; ++h) wc[h] = w_cls[o * HID + h];
        const float bc = b_cls[o];
        float uu = 0.f, accs = 0.f;
        int sp = 0;
        for (int t = 0; t < NT; ++t) {
            float logit = bc;
#pragma unroll
            for (int h = 0; h < HID; ++h)
                logit += wc[h] * ((float)pooledCnt[h * NT + t] * (1.0f / 256.0f));
            uu = sp ? logit : (uu * DECAY + logit);
            sp = (uu >= VTH) ? 1 : 0;
            accs += (float)sp;
        }
        out[b * 2 + o] = accs * (1.0f / 8.0f);
    }
}

extern "C" void kernel_launch(void* const* d_in, const int* in_sizes, int n_in,
                              void* d_out, int out_size, void* d_ws, size_t ws_size,
                              hipStream_t stream) {
    (void)in_sizes; (void)n_in; (void)d_ws; (void)ws_size; (void)out_size;
    const float* x     = (const float*)d_in[0];
    const float* w_emb = (const float*)d_in[1];
    const float* b_emb = (const float*)d_in[2];
    const float* w_ih  = (const float*)d_in[3];
    const float* w_hh  = (const float*)d_in[4];
    const float* b_ih  = (const float*)d_in[5];
    const float* b_hh  = (const float*)d_in[6];
    const float* w_cls = (const float*)d_in[7];
    const float* b_cls = (const float*)d_in[8];
    float* out = (float*)d_out;

    snn_fused_kernel<<<256, 256, 0, stream>>>(x, w_emb, b_emb, w_ih, w_hh,
                                              b_ih, b_hh, w_cls, b_cls, out);
}